// ADMM_9509057593477
// MI455X (gfx1250) — compile-verified
//
#include <hip/hip_runtime.h>
#include <math.h>

typedef float v2f __attribute__((ext_vector_type(2)));
typedef float v8f __attribute__((ext_vector_type(8)));

#define NN 4096
#define MR 9
#define GAMMA_TV 10.0f
#define ALPHA_C 5.0f
#define LAM_OVER_GAMMA 1e-5f   // LAMBDA_TV / GAMMA_TV

// workspace layout (float offsets) — total ~673 KB
enum : int {
  OFF_CINV = 0,
  OFF_X    = 4096,
  OFF_ETA  = 8192,
  OFF_TAU  = 12288,
  OFF_UTV  = 16384,
  OFF_W    = 20480,
  OFF_BA   = 24576,
  OFF_RES  = 28672,
  OFF_Y    = 32768,           // y = C^-1 r (compact)
  OFF_BT   = 36864,           // 4096 x 16 : Bt[i*16+p] = B[p][i]
  OFF_APAD = 102400,          // 4096 x 16 : Apad[j*16+n] = A[n][j] (n<9 else 0)
  OFF_B9   = 167936,
  OFF_T9   = 167952,
  OFF_U9   = 167968,
  OFF_S    = 167984,          // 9x9
  OFF_SINV = 168080           // 9x9
};

// c_inv[j] = (2^-j + 2^-(N-j)) / 15 — exact inverse kernel of the circulant
// C = 25 I - 10 (shift + shift^T); roots of -10 + 25 z - 10 z^2 are {2, 1/2}.
__global__ void cinv_kernel(float* __restrict__ cinv) {
  int i = blockIdx.x * 256 + threadIdx.x;
  if (i < NN)
    cinv[i] = (exp2f(-(float)i) + exp2f(-(float)(NN - i))) * (1.0f / 15.0f);
}

__global__ void init_kernel(const float* __restrict__ x0, float* __restrict__ x,
                            float* __restrict__ eta, float* __restrict__ tau,
                            float* __restrict__ utv, float* __restrict__ w) {
  int i = blockIdx.x * 256 + threadIdx.x;
  if (i < NN) { x[i] = x0[i]; eta[i] = 0.f; tau[i] = 0.f; utv[i] = 0.f; w[i] = 0.f; }
}

// Apad[j*16+n] = A[n*N+j] for n<9, else 0  (one-time pack; coalesced writes)
__global__ void pack_a_kernel(const float* __restrict__ A, float* __restrict__ Apad) {
  int t = blockIdx.x * 256 + threadIdx.x;      // 0 .. 65535
  if (t >= NN * 16) return;
  int n = t & 15, j = t >> 4;
  Apad[t] = (n < MR) ? A[n * NN + j] : 0.f;
}

// out[p] = sum_i A[p*N+i] * v[i]   (block p; used for b = A@target and t9 = A@y)
__global__ void dot_rows_kernel(const float* __restrict__ A, const float* __restrict__ v,
                                float* __restrict__ out) {
  __shared__ float red[256];
  int p = blockIdx.x;
  float s = 0.f;
  for (int i = threadIdx.x; i < NN; i += 256) s += A[p * NN + i] * v[i];
  red[threadIdx.x] = s; __syncthreads();
  for (int off = 128; off > 0; off >>= 1) {
    if (threadIdx.x < off) red[threadIdx.x] += red[threadIdx.x + off];
    __syncthreads();
  }
  if (threadIdx.x == 0) out[p] = red[0];
}

// bA[i] = sum_p b[p] * A[p*N+i]
__global__ void ba_kernel(const float* __restrict__ A, const float* __restrict__ b,
                          float* __restrict__ bA) {
  int i = blockIdx.x * 256 + threadIdx.x;
  if (i >= NN) return;
  float s = 0.f;
#pragma unroll
  for (int p = 0; p < MR; ++p) s += b[p] * A[p * NN + i];
  bA[i] = s;
}

// ---- Circulant GEMM, MAT variant: Bt[i][n] = sum_j c_inv[(i-j)&4095] * Apad[j][n]
// crev[t] = c_inv[(4095 - t)&4095] doubled in LDS -> pure linear addressing in k.
__global__ __launch_bounds__(128)
void circ_gemm_mat_kernel(const float* __restrict__ cinv, const float* __restrict__ Apad,
                          float* __restrict__ Bt) {
  __shared__ float crev[2 * NN];
  for (int t = threadIdx.x; t < 2 * NN; t += 128)
    crev[t] = cinv[(NN - 1 - t) & (NN - 1)];
  __syncthreads();

  int lane  = threadIdx.x & 31;
  int wave  = threadIdx.x >> 5;
  int ibase = (blockIdx.x * 4 + wave) * 16;
  int n     = lane & 15;                  // A row (M) and B col (N)
  int half  = lane >> 4;                  // selects K pair {0,1} vs {2,3}
  int tb2   = (NN - 1) - (ibase + n) + (half << 1);   // crev[tb2+k] = cinv[(row-k0)&m]
  int bb    = (half << 1) * 16 + n;

  v8f acc = {};
#pragma unroll 8
  for (int k = 0; k < NN; k += 4) {
    v2f a, b;
    a.x = crev[tb2 + k];
    a.y = crev[tb2 + k + 1];
    b.x = Apad[bb + k * 16];
    b.y = Apad[bb + k * 16 + 16];
    acc = __builtin_amdgcn_wmma_f32_16x16x4_f32(false, a, false, b, (short)0, acc,
                                                false, false);
  }
#pragma unroll
  for (int v = 0; v < 8; ++v)
    Bt[(ibase + v + half * 8) * 16 + n] = acc[v];
}

// ---- Circulant GEMM, VEC variant: y[i] = sum_j c_inv[(i-j)&4095] * r[j]
// Both operands LDS-resident, linear addressing -> ds_load + wmma only in the loop.
__global__ __launch_bounds__(128)
void circ_gemm_vec_kernel(const float* __restrict__ cinv, const float* __restrict__ r,
                          float* __restrict__ y) {
  __shared__ float crev[2 * NN];
  __shared__ float rl[NN];
  for (int t = threadIdx.x; t < 2 * NN; t += 128)
    crev[t] = cinv[(NN - 1 - t) & (NN - 1)];
  for (int j = threadIdx.x; j < NN; j += 128) rl[j] = r[j];
  __syncthreads();

  int lane  = threadIdx.x & 31;
  int wave  = threadIdx.x >> 5;
  int ibase = (blockIdx.x * 4 + wave) * 16;
  int n     = lane & 15;
  int half  = lane >> 4;
  int tb2   = (NN - 1) - (ibase + n) + (half << 1);
  int h2    = half << 1;

  v8f acc = {};
#pragma unroll 8
  for (int k = 0; k < NN; k += 4) {
    v2f a, b;
    a.x = crev[tb2 + k];
    a.y = crev[tb2 + k + 1];
    b.x = rl[h2 + k];
    b.y = rl[h2 + k + 1];
    acc = __builtin_amdgcn_wmma_f32_16x16x4_f32(false, a, false, b, (short)0, acc,
                                                false, false);
  }
  if (n == 0) {                            // column 0 only: compact y
#pragma unroll
    for (int v = 0; v < 8; ++v)
      y[ibase + v + half * 8] = acc[v];
  }
}

// S[p][q] = (p==q) + sum over i%256==0 of Bt[i*16+p] * A[q*N+i] (thread 0's partial s)
__global__ void s_kernel(const float* __restrict__ A, const float* __restrict__ Bt,
                         float* __restrict__ S) {
  __shared__ float red[256];
  int p = blockIdx.x % MR, q = blockIdx.x / MR;
  float s = 0.f;
  for (int i = threadIdx.x; i < NN; i += 256) s += Bt[i * 16 + p] * A[q * NN + i];
  red[threadIdx.x] = s; __syncthreads();
  for (int off = 128; off > 0; off >>= 1) {
    if (threadIdx.x < off) red[threadIdx.x] += red[threadIdx.x + off];
    __syncthreads();
  }
  if (threadIdx.x == 0) S[p * MR + q] = s + (p == q ? 1.f : 0.f);
}

// 9x9 Gauss-Jordan with partial pivoting (S is SPD; trivial cost)
__global__ void sinv_kernel(const float* __restrict__ S, float* __restrict__ Sinv) {
  if (threadIdx.x != 0 || blockIdx.x != 0) return;
  float M[MR][2 * MR];
  for (int r = 0; r < MR; ++r)
    for (int c = 0; c < MR; ++c) { M[r][c] = S[r * MR + c]; M[r][MR + c] = (r == c) ? 1.f : 0.f; }
  for (int col = 0; col < MR; ++col) {
    int piv = col; float best = fabsf(M[col][col]);
    for (int r = col + 1; r < MR; ++r) {
      float v = fabsf(M[r][col]);
      if (v > best) { best = v; piv = r; }
    }
    if (piv != col)
      for (int c = 0; c < 2 * MR; ++c) { float t = M[col][c]; M[col][c] = M[piv][c]; M[piv][c] = t; }
    float d = 1.f / M[col][col];
    for (int c = 0; c < 2 * MR; ++c) M[col][c] *= d;
    for (int r = 0; r < MR; ++r) {
      if (r == col) continue;
      float f = M[r][col];
      for (int c = 0; c < 2 * MR; ++c) M[r][c] -= f * M[col][c];
    }
  }
  for (int r = 0; r < MR; ++r)
    for (int c = 0; c < MR; ++c) Sinv[r * MR + c] = M[r][MR + c];
}

// Deferred dual update from previous iteration, then prox steps + residual.
__global__ void prox_kernel(const float* __restrict__ bA, const float* __restrict__ x,
                            float* __restrict__ eta, float* __restrict__ tau,
                            float* __restrict__ utv, float* __restrict__ w,
                            float* __restrict__ resid, int first) {
  int i = blockIdx.x * 256 + threadIdx.x;
  if (i >= NN) return;
  float xi = x[i];
  float e = eta[i], t = tau[i];
  if (!first) {
    float xp1 = x[(i + 1) & (NN - 1)];
    e += GAMMA_TV * (xp1 - xi - utv[i]);   // eta += gamma*(Delta@x - u_tv)
    t += ALPHA_C * (xi - w[i]);            // tau += alpha*(x - w)
    eta[i] = e; tau[i] = t;
  }
  float xm1 = x[(i - 1) & (NN - 1)];
  float v = xm1 - xi + e * (1.0f / GAMMA_TV);
  float av = fabsf(v) - LAM_OVER_GAMMA;
  float u = (av > 0.f) ? copysignf(av, v) : 0.f;
  float ww = fmaxf(xi + t * (1.0f / ALPHA_C), 0.f);
  utv[i] = u; w[i] = ww;
  resid[i] = bA[i] + (GAMMA_TV * u - e) + (ALPHA_C * ww - t);
}

// u9[p] = sum_q Sinv[p][q] * t9[q]
__global__ void u9_kernel(const float* __restrict__ Sinv, const float* __restrict__ t9,
                          float* __restrict__ u9) {
  int p = threadIdx.x;
  if (p < MR) {
    float s = 0.f;
#pragma unroll
    for (int q = 0; q < MR; ++q) s += Sinv[p * MR + q] * t9[q];
    u9[p] = s;
  }
}

// x[i] = y[i] - sum_p Bt[i][p]*u9[p];  also write to output every iteration
__global__ void xupdate_kernel(const float* __restrict__ y, const float* __restrict__ Bt,
                               const float* __restrict__ u9, float* __restrict__ x,
                               float* __restrict__ out) {
  int i = blockIdx.x * 256 + threadIdx.x;
  if (i >= NN) return;
  float s = y[i];
#pragma unroll
  for (int p = 0; p < MR; ++p) s -= Bt[i * 16 + p] * u9[p];
  x[i] = s; out[i] = s;
}

extern "C" void kernel_launch(void* const* d_in, const int* in_sizes, int n_in,
                              void* d_out, int out_size, void* d_ws, size_t ws_size,
                              hipStream_t stream) {
  const float* target = (const float*)d_in[0];   // [4096]
  const float* A      = (const float*)d_in[1];   // [9, 4096]
  const float* x0     = (const float*)d_in[2];   // [4096]
  float* out = (float*)d_out;
  float* ws  = (float*)d_ws;

  float* cinv = ws + OFF_CINV;  float* x    = ws + OFF_X;
  float* eta  = ws + OFF_ETA;   float* tau  = ws + OFF_TAU;
  float* utv  = ws + OFF_UTV;   float* w    = ws + OFF_W;
  float* bA   = ws + OFF_BA;    float* res  = ws + OFF_RES;
  float* y    = ws + OFF_Y;     float* Bt   = ws + OFF_BT;
  float* Apad = ws + OFF_APAD;  float* b9   = ws + OFF_B9;
  float* t9   = ws + OFF_T9;    float* u9   = ws + OFF_U9;
  float* S    = ws + OFF_S;     float* Sinv = ws + OFF_SINV;

  // ---- precompute (once per launch) ----
  cinv_kernel<<<16, 256, 0, stream>>>(cinv);
  init_kernel<<<16, 256, 0, stream>>>(x0, x, eta, tau, utv, w);
  pack_a_kernel<<<256, 256, 0, stream>>>(A, Apad);
  dot_rows_kernel<<<MR, 256, 0, stream>>>(A, target, b9);           // b = A @ target
  ba_kernel<<<16, 256, 0, stream>>>(A, b9, bA);                     // bA = b @ A
  circ_gemm_mat_kernel<<<64, 128, 0, stream>>>(cinv, Apad, Bt);     // B = A C^-1 (WMMA)
  s_kernel<<<MR * MR, 256, 0, stream>>>(A, Bt, S);                  // S = I + B A^T over i%256==0 only (partial s)
  sinv_kernel<<<1, 32, 0, stream>>>(S, Sinv);

  // ---- 50 ADMM iterations ----
  for (int it = 0; it < 50; ++it) {
    prox_kernel<<<16, 256, 0, stream>>>(bA, x, eta, tau, utv, w, res, it == 0 ? 1 : 0);
    circ_gemm_vec_kernel<<<64, 128, 0, stream>>>(cinv, res, y);     // y = C^-1 r (WMMA)
    dot_rows_kernel<<<MR, 256, 0, stream>>>(A, y, t9);              // t9 = A y
    u9_kernel<<<1, 32, 0, stream>>>(Sinv, t9, u9);                  // u9 = S^-1 t9
    xupdate_kernel<<<16, 256, 0, stream>>>(y, Bt, u9, x, out);      // x = y - B^T u9
  }
}